// MOTSelfAttention_64355789963775
// MI455X (gfx1250) — compile-verified
//
#include <hip/hip_runtime.h>
#include <hip/hip_bf16.h>

// ---------------------------------------------------------------------------
// MOTSelfAttention for MI455X (gfx1250).
//
// Key algebraic fact (see analysis): the reference's broadcast contraction
//   (probs[...,None] * value_layer).sum(-2)
// aligns value's Sk axis with probs' Sq axis and sums over a broadcast
// singleton, so context[b,h,s,d] = mixed_v[s,b,h*D+d] * (sum_j probs) and
// softmax rows sum to 1.  =>  output == mixed_v for ANY finite attn_mask.
// The entire l2norm/QK^T/softmax/PV pipeline (incl. the 32MB mask read) is
// dead code w.r.t. all three outputs (output, mixed_q, mixed_k).
//
// So the optimal kernel is 4 GEMMs of [1024,256]x[256,256]:
//   mixed_q = q @ Wq^T + bq            (output #2)
//   mixed_k = k @ Wk^T + bk            (output #3)
//   h       = relu(v @ Wv1^T + bv1)    (scratch)
//   output  = h @ Wv2^T + bv2          (output #1)
//
// ~0.54 GFLOP / ~7MB traffic -> memory/latency bound at 23.3 TB/s; all
// operands fit in the 192MB L2, so simple per-wave tiles are the right shape.
// fp32 WMMA (v_wmma_f32_16x16x4_f32) keeps numerics at fp32-reference level;
// no reason to drop to bf16 on a non-compute-bound problem whose GEMM results
// are directly validated outputs.
// ---------------------------------------------------------------------------

typedef __attribute__((ext_vector_type(2))) float v2f;
typedef __attribute__((ext_vector_type(8))) float v8f;

#define SQ   1024   // sequence length (M)
#define EDIM 256    // embed dim (K and N of every GEMM)

// Y[M,N] = act(X[M,K] @ W[N,K]^T + bias[N]) with M=1024, N=K=256.
// grid = (M/16, N/128); block = 256 threads = 8 waves.
// Wave w computes the 16x16 tile at (blockIdx.x*16, (blockIdx.y*8 + w)*16)
// via 64 chained V_WMMA_F32_16X16X4_F32 ops over K=256.
__global__ __launch_bounds__(256)
void gemm_f32_wmma(const float* __restrict__ X,
                   const float* __restrict__ W,
                   const float* __restrict__ bias,
                   float* __restrict__ Y,
                   int relu) {
  const int wave = threadIdx.x >> 5;
  const int lane = threadIdx.x & 31;

  const int m_base = blockIdx.x * 16;
  const int n_base = (blockIdx.y * 8 + wave) * 16;

  // 32-bit A-matrix 16x4 layout: lanes 0-15 hold M=lane, K={0,1};
  //                              lanes 16-31 hold M=lane-16, K={2,3}.
  const int mr = lane & 15;
  const int ak = (lane < 16) ? 0 : 2;
  // 32-bit B-matrix 4x16 layout: lane supplies column n = lane&15,
  // K pair {0,1} (lanes 0-15) or {2,3} (lanes 16-31).  B(k,n) = W[n][k].
  const int nc = lane & 15;
  const int bk = (lane < 16) ? 0 : 2;

  const float* __restrict__ xrow = X + (size_t)(m_base + mr) * EDIM;
  const float* __restrict__ wrow = W + (size_t)(n_base + nc) * EDIM;

  // Warm the row streams (global_prefetch_b8); rows are 1KB each.
  __builtin_prefetch(xrow, 0, 3);
  __builtin_prefetch(wrow, 0, 3);

  v8f acc = {};
#pragma unroll 16
  for (int k = 0; k < EDIM; k += 4) {
    v2f a, b;
    a.x = xrow[k + ak];
    a.y = xrow[k + ak + 1];       // pairs are contiguous -> global_load_b64
    b.x = wrow[k + bk];
    b.y = wrow[k + bk + 1];
    // (neg_a, A, neg_b, B, c_mod, C, reuse_a, reuse_b)
    acc = __builtin_amdgcn_wmma_f32_16x16x4_f32(
        false, a, false, b, (short)0, acc, false, false);
  }

  // C/D 16x16 f32 layout: lane l, VGPR j -> row = j + (l<16 ? 0 : 8), col = l&15.
  const float bn = bias[n_base + nc];
#pragma unroll
  for (int j = 0; j < 8; ++j) {
    const int mo = (lane < 16) ? j : j + 8;
    float v = acc[j] + bn;
    if (relu) v = fmaxf(v, 0.0f);
    Y[(size_t)(m_base + mo) * EDIM + n_base + nc] = v;
  }
}

extern "C" void kernel_launch(void* const* d_in, const int* in_sizes, int n_in,
                              void* d_out, int out_size, void* d_ws, size_t ws_size,
                              hipStream_t stream) {
  (void)in_sizes; (void)n_in; (void)out_size; (void)ws_size;

  const float* q   = (const float*)d_in[0];
  const float* k   = (const float*)d_in[1];
  const float* v   = (const float*)d_in[2];
  // d_in[3] = attn_mask : provably irrelevant to all outputs (see header).
  const float* Wq  = (const float*)d_in[4];
  const float* bq  = (const float*)d_in[5];
  const float* Wk  = (const float*)d_in[6];
  const float* bk  = (const float*)d_in[7];
  const float* Wv1 = (const float*)d_in[8];
  const float* bv1 = (const float*)d_in[9];
  const float* Wv2 = (const float*)d_in[10];
  const float* bv2 = (const float*)d_in[11];

  // d_out = [ output | mixed_q | mixed_k ], each S*E fp32.
  float* out = (float*)d_out;
  float* mq  = out + (size_t)SQ * EDIM;
  float* mk  = mq  + (size_t)SQ * EDIM;
  float* h   = (float*)d_ws;   // relu hidden, S*E fp32 scratch

  dim3 grid(SQ / 16, EDIM / 128);
  dim3 block(256);
  // Independent GEMMs first (q,k,v-layer1), then the dependent v-layer2.
  gemm_f32_wmma<<<grid, block, 0, stream>>>(q, Wq,  bq,  mq,  0);
  gemm_f32_wmma<<<grid, block, 0, stream>>>(k, Wk,  bk,  mk,  0);
  gemm_f32_wmma<<<grid, block, 0, stream>>>(v, Wv1, bv1, h,   1);
  gemm_f32_wmma<<<grid, block, 0, stream>>>(h, Wv2, bv2, out, 0);  // == "output"
}